// HybridMemory_63745904607628
// MI455X (gfx1250) — compile-verified
//
#include <hip/hip_runtime.h>
#include <math.h>

// ---------------- problem constants (match reference) ----------------
#define BATCH 256
#define DIM   2048
#define NROWS 65536
#define NCLUS 4096
#define TEMP  0.05f
#define MOM   0.2f
#define EPS_F 1e-6f

typedef __attribute__((ext_vector_type(2))) float v2f;
typedef __attribute__((ext_vector_type(8))) float v8f;

// ---------------- 1. histogram of labels -> nums[c] ----------------
__global__ __launch_bounds__(256) void hm_histogram(const int* __restrict__ labels,
                                                    int* __restrict__ nums) {
    int n = blockIdx.x * 256 + threadIdx.x;
    atomicAdd(&nums[labels[n]], 1);
}

// ---------------- 2. exclusive scan over NCLUS counts (1 block, 1024 thr) ----
__global__ __launch_bounds__(1024) void hm_scan(const int* __restrict__ nums,
                                                int* __restrict__ offsets) {
    __shared__ int sh[1024];
    int t = threadIdx.x;
    int c0 = t * 4;
    int v0 = nums[c0], v1 = nums[c0 + 1], v2 = nums[c0 + 2], v3 = nums[c0 + 3];
    int s = v0 + v1 + v2 + v3;
    sh[t] = s;
    __syncthreads();
    for (int off = 1; off < 1024; off <<= 1) {
        int add = (t >= off) ? sh[t - off] : 0;
        __syncthreads();
        sh[t] += add;
        __syncthreads();
    }
    int excl = sh[t] - s;
    offsets[c0]     = excl;
    offsets[c0 + 1] = excl + v0;
    offsets[c0 + 2] = excl + v0 + v1;
    offsets[c0 + 3] = excl + v0 + v1 + v2;
}

// ---------------- 3. scatter row ids into cluster-grouped order ----------------
__global__ __launch_bounds__(256) void hm_scatter(const int* __restrict__ labels,
                                                  const int* __restrict__ offsets,
                                                  int* __restrict__ cursor,
                                                  int* __restrict__ order) {
    int n = blockIdx.x * 256 + threadIdx.x;
    int c = labels[n];
    int r = atomicAdd(&cursor[c], 1);
    order[offsets[c] + r] = n;
}

// ---------------- 4. per-cluster feature sum + linear copy to output ----------
// One workgroup per cluster. Sorts its member list (deterministic sum order),
// accumulates Fsum[c,:], and streams each touched feature row to the output
// (every row belongs to exactly one cluster -> full copy in one pass).
__global__ __launch_bounds__(256) void hm_cluster_sum_copy(const float* __restrict__ features,
                                                           const int* __restrict__ order,
                                                           const int* __restrict__ offsets,
                                                           const int* __restrict__ nums,
                                                           float* __restrict__ Fsum,
                                                           float* __restrict__ outFeat) {
    __shared__ int rows[512];
    const int c = blockIdx.x;
    const int t = threadIdx.x;
    const int start = offsets[c];
    const int cnt = nums[c];
    const bool useSorted = (cnt <= 512);

    if (t == 0 && useSorted) {
        for (int i = 0; i < cnt; ++i) {
            int v = order[start + i];
            int j = i - 1;
            while (j >= 0 && rows[j] > v) { rows[j + 1] = rows[j]; --j; }
            rows[j + 1] = v;
        }
    }
    __syncthreads();

    float acc[8] = {0.f, 0.f, 0.f, 0.f, 0.f, 0.f, 0.f, 0.f};
    for (int i = 0; i < cnt; ++i) {
        int row = useSorted ? rows[i] : order[start + i];
        const float* src = features + (size_t)row * DIM;
        float* dst = outFeat + (size_t)row * DIM;
#pragma unroll
        for (int j = 0; j < 8; ++j) {
            float v = src[t + j * 256];
            acc[j] += v;
            dst[t + j * 256] = v;
        }
    }
#pragma unroll
    for (int j = 0; j < 8; ++j)
        Fsum[(size_t)c * DIM + t + j * 256] = acc[j];
}

// ---------------- 5. EMA update + L2 renorm of the 256 indexed bank rows ------
__global__ __launch_bounds__(256) void hm_ema_update(const float* __restrict__ inputs,
                                                     const float* __restrict__ features,
                                                     const int* __restrict__ indexes,
                                                     float* __restrict__ outFeat) {
    __shared__ float red[256];
    const int b = blockIdx.x;
    const int t = threadIdx.x;
    const int row = indexes[b];
    const float* f = features + (size_t)row * DIM;
    const float* in = inputs + (size_t)b * DIM;
    float v[8];
    float ss = 0.f;
#pragma unroll
    for (int j = 0; j < 8; ++j) {
        float x = MOM * f[t + j * 256] + (1.0f - MOM) * in[t + j * 256];
        v[j] = x;
        ss += x * x;
    }
    red[t] = ss;
    __syncthreads();
    for (int s = 128; s > 0; s >>= 1) {
        if (t < s) red[t] += red[t + s];
        __syncthreads();
    }
    float inv = 1.0f / sqrtf(red[0]);
    float* dst = outFeat + (size_t)row * DIM;
#pragma unroll
    for (int j = 0; j < 8; ++j)
        dst[t + j * 256] = v[j] * inv;
}

// ---------------- 6. WMMA GEMM: sim[b,c] = (inputs[b].Fsum[c]) / (TEMP*nums[c]) -
// f32 16x16x4 WMMA, each wave computes a 16(M) x 64(N) strip over K=2048.
// A frag (16x4 f32): lane L holds M=L%16, VGPR0/1 = K = (L<16 ? 0,1 : 2,3)+k0
// B frag (4x16 f32): lane L holds N=L%16, VGPR0/1 = K = (L<16 ? 0,1 : 2,3)+k0
// C/D  : element r -> M = r + 8*(L>=16), N = L%16
__global__ __launch_bounds__(256) void hm_gemm_sim_wmma(const float* __restrict__ inputs,
                                                        const float* __restrict__ Fsum,
                                                        const int* __restrict__ nums,
                                                        float* __restrict__ sim) {
    const int wave = blockIdx.x * (blockDim.x >> 5) + (threadIdx.x >> 5);
    const int lane = threadIdx.x & 31;
    const int half = lane >> 4;
    const int lm = lane & 15;
    const int mTile = wave & 15;   // 16 tiles of M
    const int nStrip = wave >> 4;  // 64 strips of 64 columns

    const float* A  = inputs + (size_t)(mTile * 16 + lm) * DIM;
    const float* B0 = Fsum + (size_t)(nStrip * 64 + lm) * DIM;
    const float* B1 = B0 + (size_t)16 * DIM;
    const float* B2 = B0 + (size_t)32 * DIM;
    const float* B3 = B0 + (size_t)48 * DIM;

    v8f acc0 = {}, acc1 = {}, acc2 = {}, acc3 = {};
    const int koff = half * 2;

#pragma unroll 8
    for (int k = 0; k < DIM; k += 4) {
        v2f a  = *(const v2f*)(A + k + koff);
        v2f b0 = *(const v2f*)(B0 + k + koff);
        v2f b1 = *(const v2f*)(B1 + k + koff);
        v2f b2 = *(const v2f*)(B2 + k + koff);
        v2f b3 = *(const v2f*)(B3 + k + koff);
        acc0 = __builtin_amdgcn_wmma_f32_16x16x4_f32(false, a, false, b0, (short)0, acc0, false, false);
        acc1 = __builtin_amdgcn_wmma_f32_16x16x4_f32(false, a, false, b1, (short)0, acc1, false, false);
        acc2 = __builtin_amdgcn_wmma_f32_16x16x4_f32(false, a, false, b2, (short)0, acc2, false, false);
        acc3 = __builtin_amdgcn_wmma_f32_16x16x4_f32(false, a, false, b3, (short)0, acc3, false, false);
    }

#pragma unroll
    for (int j = 0; j < 4; ++j) {
        v8f acc = (j == 0) ? acc0 : (j == 1) ? acc1 : (j == 2) ? acc2 : acc3;
        int n = nStrip * 64 + j * 16 + lm;
        int cnt = nums[n];
        float inv = 1.0f / (TEMP * (float)(cnt > 0 ? cnt : 1));
#pragma unroll
        for (int r = 0; r < 8; ++r) {
            int m = mTile * 16 + r + half * 8;
            sim[(size_t)m * NCLUS + n] = acc[r] * inv;
        }
    }
}

// ---------------- 7. masked softmax denominator + per-row NLL -----------------
__global__ __launch_bounds__(256) void hm_row_loss(const float* __restrict__ sim,
                                                   const int* __restrict__ nums,
                                                   const int* __restrict__ indexes,
                                                   const int* __restrict__ labels,
                                                   float* __restrict__ rowloss) {
    __shared__ float red[256];
    const int b = blockIdx.x;
    const int t = threadIdx.x;
    const float* srow = sim + (size_t)b * NCLUS;
    float lsum = 0.f;
#pragma unroll
    for (int j = 0; j < 16; ++j) {
        int c = t + j * 256;
        if (nums[c] > 0) lsum += expf(srow[c]);
    }
    red[t] = lsum;
    __syncthreads();
    for (int s = 128; s > 0; s >>= 1) {
        if (t < s) red[t] += red[t + s];
        __syncthreads();
    }
    if (t == 0) {
        int tgt = labels[indexes[b]];
        float et = (nums[tgt] > 0) ? expf(srow[tgt]) : 0.0f;
        float prob = et / (red[0] + EPS_F);
        rowloss[b] = -logf(prob + EPS_F);
    }
}

// ---------------- 8. deterministic final reduction of the loss ---------------
__global__ __launch_bounds__(256) void hm_final_loss(const float* __restrict__ rowloss,
                                                     float* __restrict__ out) {
    __shared__ float red[256];
    int t = threadIdx.x;
    red[t] = rowloss[t];
    __syncthreads();
    for (int s = 128; s > 0; s >>= 1) {
        if (t < s) red[t] += red[t + s];
        __syncthreads();
    }
    if (t == 0) out[0] = red[0] / (float)BATCH;
}

// ---------------------------------------------------------------------------
extern "C" void kernel_launch(void* const* d_in, const int* in_sizes, int n_in,
                              void* d_out, int out_size, void* d_ws, size_t ws_size,
                              hipStream_t stream) {
    const float* inputs   = (const float*)d_in[0]; // [B, D]
    const int*   indexes  = (const int*)d_in[1];   // [B]
    const float* features = (const float*)d_in[2]; // [N, D]
    const int*   labels   = (const int*)d_in[3];   // [N]
    float* out = (float*)d_out;                    // [0]=loss, [1..]=updated features

    // workspace carve-out (256B aligned regions)
    char* ws = (char*)d_ws;
    size_t off = 0;
    auto carve = [&](size_t bytes) -> void* {
        void* p = ws + off;
        off += (bytes + 255) & ~(size_t)255;
        return p;
    };
    float* Fsum    = (float*)carve((size_t)NCLUS * DIM * sizeof(float));  // 32 MB
    float* sim     = (float*)carve((size_t)BATCH * NCLUS * sizeof(float)); // 4 MB
    int*   nums    = (int*)carve((size_t)NCLUS * sizeof(int));
    int*   offsets = (int*)carve((size_t)NCLUS * sizeof(int));
    int*   cursor  = (int*)carve((size_t)NCLUS * sizeof(int));
    int*   order   = (int*)carve((size_t)NROWS * sizeof(int));
    float* rowloss = (float*)carve((size_t)BATCH * sizeof(float));
    (void)ws_size; (void)in_sizes; (void)n_in; (void)out_size;

    hipMemsetAsync(nums, 0, (size_t)NCLUS * sizeof(int), stream);
    hipMemsetAsync(cursor, 0, (size_t)NCLUS * sizeof(int), stream);

    hm_histogram<<<NROWS / 256, 256, 0, stream>>>(labels, nums);
    hm_scan<<<1, 1024, 0, stream>>>(nums, offsets);
    hm_scatter<<<NROWS / 256, 256, 0, stream>>>(labels, offsets, cursor, order);
    hm_cluster_sum_copy<<<NCLUS, 256, 0, stream>>>(features, order, offsets, nums,
                                                   Fsum, out + 1);
    hm_ema_update<<<BATCH, 256, 0, stream>>>(inputs, features, indexes, out + 1);
    hm_gemm_sim_wmma<<<128, 256, 0, stream>>>(inputs, Fsum, nums, sim);
    hm_row_loss<<<BATCH, 256, 0, stream>>>(sim, nums, indexes, labels, rowloss);
    hm_final_loss<<<1, 256, 0, stream>>>(rowloss, out);
}